// GCN_31404800868552
// MI455X (gfx1250) — compile-verified
//
#include <hip/hip_runtime.h>
#include <hip/hip_bf16.h>

typedef __attribute__((ext_vector_type(2))) float v2f;
typedef __attribute__((ext_vector_type(8))) float v8f;

// ---------------------------------------------------------------------------
// Zero fill (accumulation targets must be zeroed every call; atomics follow)
// ---------------------------------------------------------------------------
__global__ __launch_bounds__(256) void gcn_fill_zero4(float4* __restrict__ p, int n4) {
    int i = blockIdx.x * 256 + threadIdx.x;
    if (i < n4) p[i] = make_float4(0.f, 0.f, 0.f, 0.f);
}

// ---------------------------------------------------------------------------
// Tiled WMMA GEMM:  C[M,N] = f(A) @ W[K,N]
//   MODE 0: f(A) = A
//   MODE 1: f(A) = relu(A)
//   MODE 2: f(A) = relu(A) + relu(A2)   (residual fusion for layer 3)
// 256 threads = 8 wave32; wave s computes rows [blk*128 + s*16, +16) x all N.
// K tiled by 16 through LDS; V_WMMA_F32_16X16X4_F32 per ISA 7.12 layouts.
// ---------------------------------------------------------------------------
template <int K, int N, int MODE>
__global__ __launch_bounds__(256) void gcn_gemm_wmma(
        const float* __restrict__ A, const float* __restrict__ A2,
        const float* __restrict__ W, float* __restrict__ C, int M)
{
    constexpr int KB = 16;          // K block
    constexpr int NT = N / 16;      // number of 16-wide column tiles
    __shared__ __align__(16) float Ws[KB * N];        // W block  (<= 8 KB)
    __shared__ __align__(16) float As[8 * 16 * KB];   // A strips (8 KB)

    const int tid   = threadIdx.x;
    const int wave  = tid >> 5;
    const int lane  = tid & 31;
    const int lo16  = lane & 15;
    const int hi    = lane >> 4;        // 0 for lanes 0-15, 1 for lanes 16-31
    const int rowBase = blockIdx.x * 128 + wave * 16;

    v8f acc[NT] = {};

    for (int kb = 0; kb < K; kb += KB) {
        __syncthreads();
        // ---- stage W block (rows kb..kb+15, all N cols; contiguous) ----
        {
            const float4* Wg4 = (const float4*)(W + (size_t)kb * N);
            float4* Ws4 = (float4*)Ws;
            #pragma unroll
            for (int i = tid; i < (KB * N) / 4; i += 256) Ws4[i] = Wg4[i];
        }
        // ---- stage A strip block: lane handles half a row (8 floats) ----
        {
            int r  = rowBase + (lane >> 1);
            int rc = r < M ? r : (M - 1);
            int ko = kb + (lane & 1) * 8;
            const float4* Ap = (const float4*)(A + (size_t)rc * K + ko);
            float4 a0 = Ap[0], a1 = Ap[1];
            if (MODE >= 1) {
                a0.x = fmaxf(a0.x, 0.f); a0.y = fmaxf(a0.y, 0.f);
                a0.z = fmaxf(a0.z, 0.f); a0.w = fmaxf(a0.w, 0.f);
                a1.x = fmaxf(a1.x, 0.f); a1.y = fmaxf(a1.y, 0.f);
                a1.z = fmaxf(a1.z, 0.f); a1.w = fmaxf(a1.w, 0.f);
            }
            if (MODE == 2) {
                const float4* Bp = (const float4*)(A2 + (size_t)rc * K + ko);
                float4 b0 = Bp[0], b1 = Bp[1];
                a0.x += fmaxf(b0.x, 0.f); a0.y += fmaxf(b0.y, 0.f);
                a0.z += fmaxf(b0.z, 0.f); a0.w += fmaxf(b0.w, 0.f);
                a1.x += fmaxf(b1.x, 0.f); a1.y += fmaxf(b1.y, 0.f);
                a1.z += fmaxf(b1.z, 0.f); a1.w += fmaxf(b1.w, 0.f);
            }
            float4* dst = (float4*)(As + wave * 256 + (lane >> 1) * KB + (lane & 1) * 8);
            dst[0] = a0; dst[1] = a1;
        }
        __syncthreads();

        // ---- 4 WMMA k-steps over this K block ----
        #pragma unroll
        for (int kk = 0; kk < 4; ++kk) {
            const int ak = kk * 4 + hi * 2;   // K index pair for this half-wave
            v2f a;
            a.x = As[wave * 256 + lo16 * KB + ak];
            a.y = As[wave * 256 + lo16 * KB + ak + 1];
            #pragma unroll
            for (int t = 0; t < NT; ++t) {
                v2f b;
                const int bc = t * 16 + lo16;
                b.x = Ws[ak * N + bc];
                b.y = Ws[(ak + 1) * N + bc];
                acc[t] = __builtin_amdgcn_wmma_f32_16x16x4_f32(
                    false, a, false, b, (short)0, acc[t], false, false);
            }
        }
    }

    // ---- store D tiles (vgpr r: rows r / r+8 in the two half-waves) ----
    #pragma unroll
    for (int t = 0; t < NT; ++t) {
        #pragma unroll
        for (int r8 = 0; r8 < 8; ++r8) {
            int row = rowBase + r8 + hi * 8;
            if (row < M) C[(size_t)row * N + t * 16 + lo16] = acc[t][r8];
        }
    }
}

// ---------------------------------------------------------------------------
// SpMM scatter: out[rows[e]] += vals[e] * H[cols[e]]   (wave32 per edge)
// D=128 -> float4 per lane, D=64 -> float2 per lane.
// Edge indices forced scalar (s_load) via readfirstlane.
// ---------------------------------------------------------------------------
template <int D>
__global__ __launch_bounds__(256) void gcn_spmm_atomic(
        const int* __restrict__ rows, const int* __restrict__ cols,
        const float* __restrict__ vals, const float* __restrict__ H,
        float* __restrict__ out, int E)
{
    int tid  = blockIdx.x * 256 + threadIdx.x;
    int lane = tid & 31;
    int e    = __builtin_amdgcn_readfirstlane(tid >> 5);  // wave-uniform edge id
    if (e >= E) return;
    int   r = rows[e];
    int   c = cols[e];
    float v = vals[e];
    constexpr int F = D / 32;   // floats per lane
    const float* src = H   + (size_t)c * D + lane * F;
    float*       dst = out + (size_t)r * D + lane * F;
    if (F == 4) {
        float4 g = *(const float4*)src;
        atomicAdd(dst + 0, g.x * v);
        atomicAdd(dst + 1, g.y * v);
        atomicAdd(dst + 2, g.z * v);
        atomicAdd(dst + 3, g.w * v);
    } else {
        float2 g = *(const float2*)src;
        atomicAdd(dst + 0, g.x * v);
        atomicAdd(dst + 1, g.y * v);
    }
}

// ---------------------------------------------------------------------------
// Launch: GEMM1 -> SpMM1 -> GEMM2(relu) -> SpMM2 -> GEMM3(relu+relu) -> SpMM3
// ---------------------------------------------------------------------------
extern "C" void kernel_launch(void* const* d_in, const int* in_sizes, int n_in,
                              void* d_out, int out_size, void* d_ws, size_t ws_size,
                              hipStream_t stream) {
    const float* x    = (const float*)d_in[0];   // [M,512]
    const int*   rows = (const int*)  d_in[1];   // [E]
    const int*   cols = (const int*)  d_in[2];   // [E]
    const float* vals = (const float*)d_in[3];   // [E]
    const float* Win  = (const float*)d_in[4];   // [512,128]
    const float* Whid = (const float*)d_in[5];   // [128,128]
    const float* Wout = (const float*)d_in[6];   // [128,64]
    float* out = (float*)d_out;

    const int M = in_sizes[0] / 512;             // 100000
    const int E = in_sizes[1];                   // 1600000

    float* B0 = (float*)d_ws;                    // t1 / t2 / t3
    float* B1 = B0 + (size_t)M * 128;            // s1
    float* B2 = B1 + (size_t)M * 128;            // s2

    const dim3 blk(256);
    const int gemmGrid = (M + 127) / 128;
    const int spmmGrid = (E + 7) / 8;            // 8 edges (waves) per 256-thr block
    const int z128     = ((M * 128 / 4) + 255) / 256;
    const int z64      = ((M * 64 / 4) + 255) / 256;

    // layer 1: t1 = x @ Win ; s1 = spmm(t1)
    gcn_gemm_wmma<512, 128, 0><<<gemmGrid, blk, 0, stream>>>(x, nullptr, Win, B0, M);
    gcn_fill_zero4<<<z128, blk, 0, stream>>>((float4*)B1, M * 128 / 4);
    gcn_spmm_atomic<128><<<spmmGrid, blk, 0, stream>>>(rows, cols, vals, B0, B1, E);

    // layer 2: t2 = relu(s1) @ Whid ; s2 = spmm(t2)
    gcn_gemm_wmma<128, 128, 1><<<gemmGrid, blk, 0, stream>>>(B1, nullptr, Whid, B0, M);
    gcn_fill_zero4<<<z128, blk, 0, stream>>>((float4*)B2, M * 128 / 4);
    gcn_spmm_atomic<128><<<spmmGrid, blk, 0, stream>>>(rows, cols, vals, B0, B2, E);

    // layer 3: t3 = (relu(s2)+relu(s1)) @ Wout ; out = spmm(t3)
    gcn_gemm_wmma<128, 64, 2><<<gemmGrid, blk, 0, stream>>>(B2, B1, Wout, B0, M);
    gcn_fill_zero4<<<z64, blk, 0, stream>>>((float4*)out, M * 64 / 4);
    gcn_spmm_atomic<64><<<spmmGrid, blk, 0, stream>>>(rows, cols, vals, B0, out, E);
}